// HMSTA_v5_Full_7524782702617
// MI455X (gfx1250) — compile-verified
//
#include <hip/hip_runtime.h>
#include <hip/hip_bf16.h>
#include <math.h>

// ---------------- constants ----------------
#define N_NODES 100000
#define N_EDGES 600000
#define N_FEAT  166
#define HID     128

typedef float v2f __attribute__((ext_vector_type(2)));
typedef float v8f __attribute__((ext_vector_type(8)));

// fp32 WMMA: D(16x16,f32) = A(16x4,f32) x B(4x16,f32) + C
// A layout: lanes 0-15 -> M=lane, {K=0,K=1}; lanes 16-31 -> M=lane-16, {K=2,K=3}
// B layout: lanes 0-15 -> N=lane, {K=0,K=1}; lanes 16-31 -> N=lane-16, {K=2,K=3}
// C/D layout: VGPR v: lanes 0-15 -> M=v, N=lane; lanes 16-31 -> M=v+8, N=lane-16
#define WMMA_F32X4(a, b, c) \
  __builtin_amdgcn_wmma_f32_16x16x4_f32(false, (a), false, (b), (short)0, (c), false, false)

// monotonic float->uint key (ascending order preserved)
__device__ __forceinline__ unsigned fkey(float f) {
  unsigned b = __float_as_uint(f);
  return (b & 0x80000000u) ? ~b : (b | 0x80000000u);
}
__device__ __forceinline__ float fkey_inv(unsigned u) {
  unsigned b = (u & 0x80000000u) ? (u & 0x7FFFFFFFu) : ~u;
  return __uint_as_float(b);
}

__device__ __forceinline__ float bsum128(float v, float* sh) {
  int t = threadIdx.x;
  sh[t] = v; __syncthreads();
#pragma unroll
  for (int s = 64; s > 0; s >>= 1) { if (t < s) sh[t] += sh[t + s]; __syncthreads(); }
  float r = sh[0]; __syncthreads();
  return r;
}

// ---------------- init / zero ----------------
// scal[0]=tmin bits, scal[1]=tmax bits, scal[4..7]=radix-select state, scal[10]=median bits
__global__ void kInit(float* nt, float* degr, float* degh, int* upd,
                      unsigned* scal, unsigned* hist) {
  int i = blockIdx.x * blockDim.x + threadIdx.x;
  if (i < 131072) hist[i] = 0u;
  if (i < N_NODES) { nt[i] = 0.0f; upd[i] = 0; degr[i] = 1.0f; degh[i] = 1.0f; }
  if (i == 0) {
    scal[0] = 0x7F7FFFFFu; scal[1] = 0u;
    scal[4] = scal[5] = scal[6] = scal[7] = scal[10] = 0u;
  }
}

__global__ void kZeroHist(unsigned* hist) {
  int i = blockIdx.x * blockDim.x + threadIdx.x;
  if (i < 131072) hist[i] = 0u;
}

// per-edge preprocessing: nt segment-max, upd mask, hi-16 histogram of ts keys
__global__ void kEdgePre(const int* ei, const float* ts, float* nt, int* upd,
                         unsigned* hist) {
  int e = blockIdx.x * blockDim.x + threadIdx.x;
  if (e >= N_EDGES) return;
  float t = ts[e];
  int r = ei[e], c = ei[N_EDGES + e];
  atomicAdd(&hist[fkey(t) >> 16], 1u);
  // timestamps are >= 0 -> uint bit compare == float compare
  atomicMax((unsigned*)(nt + c), __float_as_uint(t));
  upd[r] = 1; upd[c] = 1;
}

__global__ void kMinMaxNt(const float* nt, unsigned* scal) {
  int i = blockIdx.x * blockDim.x + threadIdx.x;
  if (i >= N_NODES) return;
  unsigned b = __float_as_uint(nt[i]);  // nt >= 0
  atomicMin(&scal[0], b);
  atomicMax(&scal[1], b);
}

// ---------------- exact median via 2-level radix select ----------------
__global__ void kSelHi(const unsigned* hist, unsigned* scal) {
  if (threadIdx.x | blockIdx.x) return;
  const unsigned k1 = (N_EDGES / 2) - 1, k2 = N_EDGES / 2;
  unsigned cum = 0; int f1 = 0, f2 = 0;
  for (unsigned bin = 0; bin < 65536u; ++bin) {
    unsigned cnt = hist[bin];
    if (!f1 && k1 < cum + cnt) { scal[4] = bin; scal[5] = k1 - cum; f1 = 1; }
    if (!f2 && k2 < cum + cnt) { scal[6] = bin; scal[7] = k2 - cum; f2 = 1; }
    cum += cnt;
    if (f1 && f2) break;
  }
}

__global__ void kHistLo(const float* ts, const unsigned* scal, unsigned* hist) {
  int e = blockIdx.x * blockDim.x + threadIdx.x;
  if (e >= N_EDGES) return;
  unsigned u = fkey(ts[e]);
  unsigned hi = u >> 16, lo = u & 0xFFFFu;
  if (hi == scal[4]) atomicAdd(&hist[lo], 1u);
  if (hi == scal[6]) atomicAdd(&hist[65536 + lo], 1u);
}

__global__ void kSelLo(const unsigned* hist, unsigned* scal) {
  if (threadIdx.x | blockIdx.x) return;
  float v[2];
  for (int i = 0; i < 2; ++i) {
    unsigned hi = scal[4 + 2 * i], rem = scal[5 + 2 * i];
    const unsigned* hh = hist + (i ? 65536 : 0);
    unsigned cum = 0, lo = 0;
    for (unsigned bin = 0; bin < 65536u; ++bin) {
      unsigned cnt = hh[bin];
      if (rem < cum + cnt) { lo = bin; break; }
      cum += cnt;
    }
    v[i] = fkey_inv((hi << 16) | lo);
  }
  scal[10] = __float_as_uint(0.5f * (v[0] + v[1]));
}

// ---------------- input projection (WMMA, K=166 = 41 full chunks + peeled tail) ----
// h = relu(x @ W_in.T + b_in + nt_norm*W_time + b_time) + memory
// Full chunks k=0..160 are unpredicated (kk max = 163 < 166 for both lane halves).
// Tail K=164,165: lanes 16-31 would address K=166/167 (OOB on the final rows of
// x / W_in), so they read a safe address (k=0) and multiply by 0 -> branch-free,
// EXEC stays all-ones (required for WMMA).
__global__ void kProj(const float* __restrict__ x, const float* __restrict__ Win,
                      const float* __restrict__ bin, const float* __restrict__ Wt,
                      const float* __restrict__ bt, const float* __restrict__ mem,
                      const float* __restrict__ nt, const unsigned* __restrict__ scal,
                      float* __restrict__ h) {
  int lane = threadIdx.x & 31, w = threadIdx.x >> 5;
  int row0 = blockIdx.x * 16, col0 = w * 16;
  int rA = row0 + (lane & 15), nB = col0 + (lane & 15);
  int khalf = (lane >> 4) << 1;
  const float* xrow = x + rA * N_FEAT;
  const float* wrow = Win + nB * N_FEAT;
  v8f c = {};
  for (int k = 0; k < 164; k += 4) {
    int kk = k + khalf;
    v2f a, b;
    a.x = xrow[kk]; a.y = xrow[kk + 1];
    b.x = wrow[kk]; b.y = wrow[kk + 1];
    c = WMMA_F32X4(a, b, c);
  }
  {
    float msk = (khalf == 0) ? 1.0f : 0.0f;
    int kk = (khalf == 0) ? 164 : 0;
    v2f a, b;
    a.x = xrow[kk] * msk; a.y = xrow[kk + 1] * msk;
    b.x = wrow[kk] * msk; b.y = wrow[kk + 1] * msk;
    c = WMMA_F32X4(a, b, c);
  }
  float tmin = __uint_as_float(scal[0]), tmax = __uint_as_float(scal[1]);
  int col = col0 + (lane & 15);
  int rbase = row0 + ((lane >> 4) << 3);
#pragma unroll
  for (int v = 0; v < 8; ++v) {
    int row = rbase + v;
    float ntv = nt[row];
    float ntn = (tmax > tmin) ? (ntv - tmin) / (tmax - tmin + 1e-8f) : ntv;
    float val = c[v] + bin[col] + ntn * Wt[col] + bt[col];
    val = fmaxf(val, 0.0f) + mem[row * HID + col];
    h[row * HID + col] = val;
  }
}

// ---------------- fused GRU (6 WMMA tiles/wave), writes new_memory ----------------
__global__ void kGru(const float* __restrict__ h, const float* __restrict__ mem,
                     const float* __restrict__ Wih, const float* __restrict__ Whh,
                     const float* __restrict__ bih, const float* __restrict__ bhh,
                     const int* __restrict__ upd, float* __restrict__ nm) {
  int lane = threadIdx.x & 31, w = threadIdx.x >> 5;
  int row0 = blockIdx.x * 16, cg = w * 16;
  int rA = row0 + (lane & 15), nb = cg + (lane & 15);
  int khalf = (lane >> 4) << 1;
  v8f ci[3] = {}, ch[3] = {};
  for (int k = 0; k < HID; k += 4) {
    int kk = k + khalf;
    v2f ah, am;
    ah.x = h[rA * HID + kk];   ah.y = h[rA * HID + kk + 1];
    am.x = mem[rA * HID + kk]; am.y = mem[rA * HID + kk + 1];
#pragma unroll
    for (int s = 0; s < 3; ++s) {
      int n = s * HID + nb;
      v2f bi, bh;
      bi.x = Wih[n * HID + kk]; bi.y = Wih[n * HID + kk + 1];
      bh.x = Whh[n * HID + kk]; bh.y = Whh[n * HID + kk + 1];
      ci[s] = WMMA_F32X4(ah, bi, ci[s]);
      ch[s] = WMMA_F32X4(am, bh, ch[s]);
    }
  }
  int col = cg + (lane & 15);
  int rbase = row0 + ((lane >> 4) << 3);
#pragma unroll
  for (int v = 0; v < 8; ++v) {
    int row = rbase + v;
    float ir = ci[0][v] + bih[col],           hr = ch[0][v] + bhh[col];
    float iz = ci[1][v] + bih[HID + col],     hz = ch[1][v] + bhh[HID + col];
    float ic = ci[2][v] + bih[2 * HID + col], hc = ch[2][v] + bhh[2 * HID + col];
    float r = 1.0f / (1.0f + expf(-(ir + hr)));
    float z = 1.0f / (1.0f + expf(-(iz + hz)));
    float cc = tanhf(ic + r * hc);
    float m = mem[row * HID + col];
    nm[row * HID + col] = upd[row] ? (1.0f - z) * cc + z * m : m;
  }
}

// ---------------- generic [*,128] @ W[N,128].T (+bias, optional relu) ----------------
// blockDim.x = 2*N (N/16 waves); N in {64,128}
__global__ void kLin(const float* __restrict__ A, const float* __restrict__ W,
                     const float* __restrict__ bias, float* __restrict__ out,
                     int N, int doRelu) {
  int lane = threadIdx.x & 31, w = threadIdx.x >> 5;
  int row0 = blockIdx.x * 16, col0 = w * 16;
  int rA = row0 + (lane & 15), nB = col0 + (lane & 15);
  int khalf = (lane >> 4) << 1;
  v8f c = {};
  for (int k = 0; k < HID; k += 4) {
    int kk = k + khalf;
    v2f a, b;
    a.x = A[rA * HID + kk]; a.y = A[rA * HID + kk + 1];
    b.x = W[nB * HID + kk]; b.y = W[nB * HID + kk + 1];
    c = WMMA_F32X4(a, b, c);
  }
  int col = col0 + (lane & 15);
  int rbase = row0 + ((lane >> 4) << 3);
#pragma unroll
  for (int v = 0; v < 8; ++v) {
    float val = c[v] + (bias ? bias[col] : 0.0f);
    if (doRelu) val = fmaxf(val, 0.0f);
    out[(rbase + v) * N + col] = val;
  }
}

// ---------------- GCN scatter phase ----------------
__global__ void kDeg(const int* ei, const float* ts, const unsigned* scal,
                     float* degr, float* degh) {
  int e = blockIdx.x * blockDim.x + threadIdx.x;
  if (e >= N_EDGES) return;
  int c = ei[N_EDGES + e];
  float med = __uint_as_float(scal[10]);
  if (ts[e] >= med) atomicAdd(&degr[c], 1.0f);
  else              atomicAdd(&degh[c], 1.0f);
}

__global__ void kAggInit(const float* hlr, const float* hlh,
                         const float* degr, const float* degh,
                         float* aggr, float* aggh) {
  int i = blockIdx.x * blockDim.x + threadIdx.x;
  if (i >= N_NODES * HID) return;
  int n = i >> 7;
  aggr[i] = hlr[i] / degr[n];   // self-loop term: hl * dis^2
  aggh[i] = hlh[i] / degh[n];
}

__global__ void kEdgeAgg(const int* __restrict__ ei, const float* __restrict__ ts,
                         const unsigned* __restrict__ scal,
                         const float* __restrict__ degr, const float* __restrict__ degh,
                         const float* __restrict__ hlr, const float* __restrict__ hlh,
                         float* __restrict__ aggr, float* __restrict__ aggh) {
  int e = blockIdx.x, c = threadIdx.x;
  int r = ei[e], cl = ei[N_EDGES + e];
  float med = __uint_as_float(scal[10]);
  if (ts[e] >= med) {
    float coef = rsqrtf(degr[r]) * rsqrtf(degr[cl]);
    atomicAdd(&aggr[cl * HID + c], hlr[r * HID + c] * coef);
  } else {
    float coef = rsqrtf(degh[r]) * rsqrtf(degh[cl]);
    atomicAdd(&aggh[cl * HID + c], hlh[r * HID + c] * coef);
  }
}

// ---------------- multi-path softmax combine + layernorm ----------------
__global__ void kCombine(const float* __restrict__ aggr, const float* __restrict__ aggh,
                         const float* __restrict__ bgr, const float* __restrict__ bgh,
                         const float* __restrict__ g, const float* __restrict__ b,
                         float* __restrict__ h2) {
  __shared__ float sh[HID];
  int n = blockIdx.x, c = threadIdx.x;
  float hr = aggr[n * HID + c] + bgr[c];
  float hh = aggh[n * HID + c] + bgh[c];
  float mr = bsum128(hr, sh) * (1.0f / HID);
  float mh = bsum128(hh, sh) * (1.0f / HID);
  float mm = fmaxf(mr, mh);
  float e0 = expf(mr - mm), e1 = expf(mh - mm);
  float inv = 1.0f / (e0 + e1);
  float s = (e0 * inv) * hr + (e1 * inv) * hh;
  float mean = bsum128(s, sh) * (1.0f / HID);
  float d = s - mean;
  float var = bsum128(d * d, sh) * (1.0f / HID);
  h2[n * HID + c] = d * rsqrtf(var + 1e-5f) * g[c] + b[c];
}

// ---------------- residual + layernorm (attention) ----------------
__global__ void kLnAtt(const float* __restrict__ h2, const float* __restrict__ att,
                       const float* __restrict__ g, const float* __restrict__ b,
                       float* __restrict__ h3) {
  __shared__ float sh[HID];
  int n = blockIdx.x, c = threadIdx.x;
  float s = h2[n * HID + c] + att[n * HID + c];
  float mean = bsum128(s, sh) * (1.0f / HID);
  float d = s - mean;
  float var = bsum128(d * d, sh) * (1.0f / HID);
  h3[n * HID + c] = d * rsqrtf(var + 1e-5f) * g[c] + b[c];
}

// ---------------- classifier head ----------------
__global__ void kLogits(const float* __restrict__ t64, const float* __restrict__ W2,
                        const float* __restrict__ b2, float* __restrict__ out) {
  int n = blockIdx.x * blockDim.x + threadIdx.x;
  if (n >= N_NODES) return;
  float a0 = b2[0], a1 = b2[1];
#pragma unroll 8
  for (int k = 0; k < 64; ++k) {
    float t = t64[n * 64 + k];
    a0 += t * W2[k];
    a1 += t * W2[64 + k];
  }
  out[n * 2] = a0;
  out[n * 2 + 1] = a1;
}

// ---------------- host launcher ----------------
extern "C" void kernel_launch(void* const* d_in, const int* in_sizes, int n_in,
                              void* d_out, int out_size, void* d_ws, size_t ws_size,
                              hipStream_t stream) {
  (void)in_sizes; (void)n_in; (void)out_size; (void)ws_size;
  const float* x     = (const float*)d_in[0];
  const int*   ei    = (const int*)d_in[1];
  const float* ts    = (const float*)d_in[2];
  const float* mem   = (const float*)d_in[3];
  const float* Win   = (const float*)d_in[4];
  const float* bin   = (const float*)d_in[5];
  const float* Wt    = (const float*)d_in[6];
  const float* bt    = (const float*)d_in[7];
  const float* Wih   = (const float*)d_in[8];
  const float* Whh   = (const float*)d_in[9];
  const float* bih   = (const float*)d_in[10];
  const float* bhh   = (const float*)d_in[11];
  const float* Wgr   = (const float*)d_in[12];
  const float* bgr   = (const float*)d_in[13];
  const float* Wgh   = (const float*)d_in[14];
  const float* bgh   = (const float*)d_in[15];
  const float* lnpg  = (const float*)d_in[16];
  const float* lnpb  = (const float*)d_in[17];
  const float* Wv    = (const float*)d_in[18];
  const float* bv    = (const float*)d_in[19];
  const float* Wo    = (const float*)d_in[20];
  const float* bo    = (const float*)d_in[21];
  const float* lnag  = (const float*)d_in[22];
  const float* lnab  = (const float*)d_in[23];
  const float* W1    = (const float*)d_in[24];
  const float* b1    = (const float*)d_in[25];
  const float* W2    = (const float*)d_in[26];
  const float* b2    = (const float*)d_in[27];

  float* out_logits = (float*)d_out;                 // [N,2]
  float* out_newmem = out_logits + N_NODES * 2;      // [N,128]

  // workspace layout (floats); peak ~310 MB with buffer reuse
  float*    ws   = (float*)d_ws;
  float*    nt   = ws;                                // 100000
  float*    degr = ws + 100000;                       // 100000
  float*    degh = ws + 200000;                       // 100000
  int*      upd  = (int*)(ws + 300000);               // 100000
  unsigned* scal = (unsigned*)(ws + 400000);          // 64
  unsigned* hist = (unsigned*)(ws + 400064);          // 131072
  float* h    = ws + 600000;                          // 12.8M
  float* hlr  = ws + 13400000;
  float* hlh  = ws + 26200000;
  float* aggr = ws + 39000000;
  float* aggh = ws + 51800000;
  float* h2   = ws + 64600000;
  float* tbuf = hlr;   // reuse after GCN scatter done
  float* att  = hlh;
  float* t64  = aggr;
  float* h3   = h;

  const int RT = N_NODES / 16;                        // 6250 row tiles (exact)
  const int EB = (N_EDGES + 255) / 256;
  const int NB = (N_NODES + 255) / 256;

  // phase 0: init, edge preprocessing, nt min/max, exact median
  kInit<<<(131072 + 255) / 256, 256, 0, stream>>>(nt, degr, degh, upd, scal, hist);
  kEdgePre<<<EB, 256, 0, stream>>>(ei, ts, nt, upd, hist);
  kMinMaxNt<<<NB, 256, 0, stream>>>(nt, scal);
  kSelHi<<<1, 1, 0, stream>>>(hist, scal);
  kZeroHist<<<(131072 + 255) / 256, 256, 0, stream>>>(hist);
  kHistLo<<<EB, 256, 0, stream>>>(ts, scal, hist);
  kSelLo<<<1, 1, 0, stream>>>(hist, scal);

  // phase 1: input projection + temporal encoding (WMMA)
  kProj<<<RT, 256, 0, stream>>>(x, Win, bin, Wt, bt, mem, nt, scal, h);

  // phase 2: fused GRU memory update -> out_newmem (WMMA x6 per wave)
  kGru<<<RT, 256, 0, stream>>>(h, mem, Wih, Whh, bih, bhh, upd, out_newmem);

  // phase 3: GCN linears (WMMA), degree, scatter aggregation
  kLin<<<RT, 256, 0, stream>>>(h, Wgr, nullptr, hlr, HID, 0);
  kLin<<<RT, 256, 0, stream>>>(h, Wgh, nullptr, hlh, HID, 0);
  kDeg<<<EB, 256, 0, stream>>>(ei, ts, scal, degr, degh);
  kAggInit<<<(N_NODES * HID + 255) / 256, 256, 0, stream>>>(hlr, hlh, degr, degh, aggr, aggh);
  kEdgeAgg<<<N_EDGES, HID, 0, stream>>>(ei, ts, scal, degr, degh, hlr, hlh, aggr, aggh);

  // phase 4: path softmax + LN
  kCombine<<<N_NODES, HID, 0, stream>>>(aggr, aggh, bgr, bgh, lnpg, lnpb, h2);

  // phase 5: anomaly attention (query path cancels exactly) + LN (WMMA x2)
  kLin<<<RT, 256, 0, stream>>>(h2, Wv, bv, tbuf, HID, 0);
  kLin<<<RT, 256, 0, stream>>>(tbuf, Wo, bo, att, HID, 0);
  kLnAtt<<<N_NODES, HID, 0, stream>>>(h2, att, lnag, lnab, h3);

  // phase 6: classifier (WMMA for 128->64, tiny 64->2 scalar)
  kLin<<<RT, 128, 0, stream>>>(h3, W1, b1, t64, 64, 1);
  kLogits<<<NB, 256, 0, stream>>>(t64, W2, b2, out_logits);
}